// NeRFRenderer_86921548137094
// MI455X (gfx1250) — compile-verified
//
#include <hip/hip_runtime.h>
#include <hip/hip_bf16.h>
#include <stdint.h>

#define N_SAMPLES      64
#define RAYS_PER_BLOCK 64          // 2 waves (wave32); 2^20 rays divides evenly
#define ROW_STRIDE     68          // floats: 272B rows, 16B-aligned, low bank conflict
#define FAR_DELTA      1e10f
#define EPS_T          1e-10f

// Clang-native 4-float vector (accepted by __builtin_nontemporal_load/store,
// and gives ds_load_b128 from LDS).
typedef __attribute__((ext_vector_type(4))) float v4f;

// CDNA5 async global->LDS copy (ASYNCcnt-tracked). Inline asm because the
// clang builtin arity differs between toolchains. vdst = LDS byte offset
// (addr[31:0] of a generic LDS pointer IS the wave-relative LDS offset),
// vaddr = 64-bit global address, 'off' = SADDR NULL (GV mode).
__device__ __forceinline__ void async_ld_b128(uint32_t lds_off, const void* gp) {
    uint64_t ga = (uint64_t)(uintptr_t)gp;
    asm volatile("global_load_async_to_lds_b128 %0, %1, off"
                 :: "v"(lds_off), "v"(ga) : "memory");
}

__device__ __forceinline__ void wait_async_zero() {
    asm volatile("s_wait_asynccnt 0" ::: "memory");
}

__global__ __launch_bounds__(RAYS_PER_BLOCK)
void nerf_composite_kernel(const float* __restrict__ sigmas,
                           const float* __restrict__ rgbs,
                           const float* __restrict__ z_vals,
                           float* __restrict__ out_rgb,
                           float* __restrict__ out_depth,
                           float* __restrict__ out_trans,
                           int n_rays)
{
    __shared__ float sm_sig[RAYS_PER_BLOCK * ROW_STRIDE];
    __shared__ float sm_z  [RAYS_PER_BLOCK * ROW_STRIDE];

    const int t    = threadIdx.x;
    const int ray0 = blockIdx.x * RAYS_PER_BLOCK;

    // LDS byte offsets of the staging buffers (low 32 bits of generic ptr).
    const uint32_t sig_base = (uint32_t)(uintptr_t)(&sm_sig[0]);
    const uint32_t z_base   = (uint32_t)(uintptr_t)(&sm_z[0]);

    // ---- Stage sigma & z rows with coalesced async b128 copies ----
    // 64 rays * 16 float4 chunks per array; thread t takes chunks
    // t, t+64, ... so consecutive lanes touch consecutive 16B chunks.
    // Uniform full-block test hoisted out so the hot path issues all 32
    // async copies back-to-back with no exec-mask churn.
    const float* gsig = sigmas + (size_t)ray0 * N_SAMPLES;
    const float* gz   = z_vals + (size_t)ray0 * N_SAMPLES;
    if (ray0 + RAYS_PER_BLOCK <= n_rays) {
#pragma unroll
        for (int i = 0; i < 16; ++i) {
            int c  = t + i * RAYS_PER_BLOCK;               // chunk id 0..1023
            uint32_t loff = (uint32_t)((c >> 4) * (ROW_STRIDE * 4) + (c & 15) * 16);
            async_ld_b128(sig_base + loff, gsig + (size_t)c * 4);
            async_ld_b128(z_base   + loff, gz   + (size_t)c * 4);
        }
    } else {                                               // ragged tail block
        for (int i = 0; i < 16; ++i) {
            int c  = t + i * RAYS_PER_BLOCK;
            if (ray0 + (c >> 4) < n_rays) {
                uint32_t loff = (uint32_t)((c >> 4) * (ROW_STRIDE * 4) + (c & 15) * 16);
                async_ld_b128(sig_base + loff, gsig + (size_t)c * 4);
                async_ld_b128(z_base   + loff, gz   + (size_t)c * 4);
            }
        }
    }
    wait_async_zero();        // my wave's async copies have landed in LDS
    __syncthreads();          // everyone's copies visible to everyone

    const int ray = ray0 + t;
    if (ray >= n_rays) return;

    const float* __restrict__ srow = &sm_sig[t * ROW_STRIDE];
    const float* __restrict__ zrow = &sm_z  [t * ROW_STRIDE];
    const v4f* __restrict__ rgbq =
        (const v4f*)(rgbs + (size_t)ray * (3 * N_SAMPLES));

    float T = 1.0f, Tlast = 1.0f;
    float accR = 0.f, accG = 0.f, accB = 0.f, accD = 0.f, accW = 0.f;

    for (int c = 0; c < 16; ++c) {
        // 12 floats = RGB for samples 4c..4c+3, streamed non-temporally.
        v4f q0 = __builtin_nontemporal_load(rgbq + c * 3 + 0);
        v4f q1 = __builtin_nontemporal_load(rgbq + c * 3 + 1);
        v4f q2 = __builtin_nontemporal_load(rgbq + c * 3 + 2);
        // sigma + z for this chunk as b128 LDS reads (rows are 16B aligned).
        v4f sq = *(const v4f*)&srow[c * 4];
        v4f zq = *(const v4f*)&zrow[c * 4];
        float znf = (c < 15) ? zrow[c * 4 + 4] : zq.w;     // first z of next chunk

        const float cr[4]  = {q0.x, q0.w, q1.z, q2.y};
        const float cg[4]  = {q0.y, q1.x, q1.w, q2.z};
        const float cb[4]  = {q0.z, q1.y, q2.x, q2.w};
        const float sg4[4] = {sq.x, sq.y, sq.z, sq.w};
        const float zc4[4] = {zq.x, zq.y, zq.z, zq.w};
        const float zn4[4] = {zq.y, zq.z, zq.w, znf};
#pragma unroll
        for (int j = 0; j < 4; ++j) {
            int s = c * 4 + j;
            float delta = (s < N_SAMPLES - 1) ? (zn4[j] - zc4[j]) : FAR_DELTA;
            float sg = sg4[j] > 0.0f ? sg4[j] : 0.0f;      // relu
            float alpha = 1.0f - expf(-sg * delta);
            if (s == N_SAMPLES - 1) Tlast = T;             // accum_trans[:, -1]
            float w = alpha * T;
            accR += w * cr[j];
            accG += w * cg[j];
            accB += w * cb[j];
            accD += w * zc4[j];
            accW += w;
            T *= (1.0f - alpha + EPS_T);
        }
    }

    float bg = 1.0f - accW;   // white background compositing
    __builtin_nontemporal_store(accR + bg, &out_rgb[3 * (size_t)ray + 0]);
    __builtin_nontemporal_store(accG + bg, &out_rgb[3 * (size_t)ray + 1]);
    __builtin_nontemporal_store(accB + bg, &out_rgb[3 * (size_t)ray + 2]);
    __builtin_nontemporal_store(accD,      &out_depth[ray]);
    __builtin_nontemporal_store(Tlast,     &out_trans[ray]);
}

extern "C" void kernel_launch(void* const* d_in, const int* in_sizes, int n_in,
                              void* d_out, int out_size, void* d_ws, size_t ws_size,
                              hipStream_t stream) {
    (void)n_in; (void)out_size; (void)d_ws; (void)ws_size;
    const float* sigmas = (const float*)d_in[0];   // (N, 64)
    const float* rgbs   = (const float*)d_in[1];   // (N, 64, 3)
    const float* z_vals = (const float*)d_in[2];   // (N, 64)

    const int n_rays = in_sizes[0] / N_SAMPLES;

    float* out      = (float*)d_out;
    float* out_rgb  = out;                              // (N,3) flat
    float* out_dep  = out + (size_t)3 * n_rays;         // (N,)
    float* out_tr   = out + (size_t)4 * n_rays;         // (N,)

    const int blocks = (n_rays + RAYS_PER_BLOCK - 1) / RAYS_PER_BLOCK;
    nerf_composite_kernel<<<blocks, RAYS_PER_BLOCK, 0, stream>>>(
        sigmas, rgbs, z_vals, out_rgb, out_dep, out_tr, n_rays);
}